// VQEmbedding_48816598286645
// MI455X (gfx1250) — compile-verified
//
#include <hip/hip_runtime.h>
#include <hip/hip_bf16.h>

typedef __attribute__((ext_vector_type(2))) float v2f;
typedef __attribute__((ext_vector_type(8))) float v8f;
typedef __attribute__((ext_vector_type(2))) int   v2i;

#define KCODES 512
#define DDIM   128
#define SPAT   32768          // 32*32*32 spatial positions per batch
#define NROWS  262144         // 8 * SPAT
#define NCHUNK 33             // 32 K-chunks of emb + 1 e2 chunk

#define AS1 __attribute__((address_space(1)))
#define AS3 __attribute__((address_space(3)))

#if __has_builtin(__builtin_amdgcn_global_load_async_to_lds_b64) && \
    __has_builtin(__builtin_amdgcn_s_wait_asynccnt)
#define USE_ASYNC_STAGING 1
#else
#define USE_ASYNC_STAGING 0
#endif

// ---------------------------------------------------------------------------
// Kernel 0: e2[k] = ||emb_k||^2   (512 values into d_ws)
// ---------------------------------------------------------------------------
__global__ void vq_e2_kernel(const float* __restrict__ emb,
                             float* __restrict__ e2) {
    int k = blockIdx.x * blockDim.x + threadIdx.x;
    if (k < KCODES) {
        const float* r = emb + k * DDIM;
        float s = 0.0f;
        #pragma unroll 8
        for (int d = 0; d < DDIM; ++d) s += r[d] * r[d];
        e2[k] = s;
    }
}

// ---------------------------------------------------------------------------
// Kernel 1: fused (-2 z) @ emb^T + ||e||^2 GEMM via V_WMMA_F32_16X16X4_F32,
// running argmin over the 512 codewords.
//
// LDS: [32 col-tiles][33 chunks][32 lanes] float2 B-fragments (264 KB).
// Chunk 32 is the "e2 chunk" (B[0][n] = ||e_n||^2), so accumulator init is
// one WMMA against inline-0 C — no global traffic in the hot loop.
// Each wave runs TWO 16-row M-tiles sharing every B-fragment: 1 ds_load_b64
// per 2 WMMAs, two independent accumulation chains.
//
// 264 KB LDS => exactly one block per WGP (2 waves/SIMD), so launch_bounds
// (256, 1) lets the compiler keep both A-panels (~128 VGPRs) + accumulators
// in registers instead of spilling to scratch.
// ---------------------------------------------------------------------------
__global__ void __launch_bounds__(256, 1)
vq_argmin_kernel(const float* __restrict__ z,
                 const float* __restrict__ emb,
                 const float* __restrict__ e2g,
                 int* __restrict__ out) {
    extern __shared__ v2f ldsB[];   // 32*33*32 float2 = 264 KB

    const int tid  = threadIdx.x;
    const int wave = tid >> 5;
    const int lane = tid & 31;
    const int lmod = lane & 15;
    const int lhi  = lane >> 4;

    // ---- stage emb^T into LDS in WMMA B-fragment order --------------------
#if USE_ASYNC_STAGING
    for (int e = tid; e < 32 * 32 * 32; e += 256) {
        int t  = e >> 10;
        int c  = (e >> 5) & 31;
        int ln = e & 31;
        int kc = (t << 4) + (ln & 15);
        int d  = (c << 2) + ((ln >> 4) << 1);
        const float* g = emb + kc * DDIM + d;
        v2f* l = &ldsB[(t * NCHUNK + c) * 32 + ln];
        __builtin_amdgcn_global_load_async_to_lds_b64(
            (AS1 v2i*)g, (AS3 v2i*)l, 0, 0);
    }
    // e2 chunk via plain DS stores (needs gather + zero fill)
    for (int e = tid; e < 32 * 32; e += 256) {
        int t  = e >> 5;
        int ln = e & 31;
        v2f v;
        v.x = (ln < 16) ? e2g[(t << 4) + ln] : 0.0f;
        v.y = 0.0f;
        ldsB[(t * NCHUNK + 32) * 32 + ln] = v;
    }
    __builtin_amdgcn_s_wait_asynccnt(0);
    __syncthreads();
#else
    for (int e = tid; e < 32 * NCHUNK * 32; e += 256) {
        int t   = e / (NCHUNK * 32);
        int rem = e - t * (NCHUNK * 32);
        int c   = rem >> 5;
        int ln  = rem & 31;
        v2f v;
        if (c < 32) {
            int kc = (t << 4) + (ln & 15);
            int d  = (c << 2) + ((ln >> 4) << 1);
            v = *(const v2f*)(emb + kc * DDIM + d);
        } else {
            v.x = (ln < 16) ? e2g[(t << 4) + ln] : 0.0f;
            v.y = 0.0f;
        }
        ldsB[e] = v;
    }
    __syncthreads();
#endif

    // A-fragment for the e2 chunk: A[m][0] = 1, other K-slots 0
    v2f aE;
    aE.x = (lane < 16) ? 1.0f : 0.0f;
    aE.y = 0.0f;

    // ---- each wave: 2 pairs of M-tiles (64 rows total) --------------------
    for (int p = 0; p < 2; ++p) {
        const int tile0 = blockIdx.x * 32 + wave * 4 + 2 * p;
        const int n0    = tile0 << 4;          // rows n0..n0+31 (same batch)
        const int b     = n0 >> 15;
        const int s0    = n0 & (SPAT - 1);

        // A-panels for both 16-row tiles: A[m][d] = -2 * z[b, d, s0 + m]
        const float* zb = z + ((size_t)b * DDIM) * SPAT + s0 + lmod;
        const int dhalf = lhi << 1;
        v2f a0[32], a1[32];
        #pragma unroll
        for (int c = 0; c < 32; ++c) {
            int d0 = (c << 2) + dhalf;
            const float* p0 = zb + (size_t)d0 * SPAT;
            const float* p1 = p0 + (size_t)SPAT;
            v2f x0, x1;
            x0.x = -2.0f * p0[0];  x0.y = -2.0f * p1[0];
            x1.x = -2.0f * p0[16]; x1.y = -2.0f * p1[16];
            a0[c] = x0;
            a1[c] = x1;
        }

        float bs0[8], bs1[8];
        int   bi0[8], bi1[8];
        #pragma unroll
        for (int r = 0; r < 8; ++r) {
            bs0[r] = __builtin_inff(); bi0[r] = 0;
            bs1[r] = __builtin_inff(); bi1[r] = 0;
        }

        // ---- loop over 32 column tiles of 16 codewords -------------------
        for (int t = 0; t < 32; ++t) {
            const v2f* bt = ldsB + t * (NCHUNK * 32) + lane;

            // accumulator init = broadcast ||e_n||^2 via the e2 chunk
            v2f bE = bt[32 << 5];
            v8f zero8 = {};
            v8f acc0 = __builtin_amdgcn_wmma_f32_16x16x4_f32(
                false, aE, false, bE, (short)0, zero8, false, false);
            v8f acc1 = __builtin_amdgcn_wmma_f32_16x16x4_f32(
                false, aE, false, bE, (short)0, zero8, false, false);

            #pragma unroll
            for (int c = 0; c < 32; ++c) {
                v2f bf = bt[c << 5];
                acc0 = __builtin_amdgcn_wmma_f32_16x16x4_f32(
                    false, a0[c], false, bf, (short)0, acc0, false, false);
                acc1 = __builtin_amdgcn_wmma_f32_16x16x4_f32(
                    false, a1[c], false, bf, (short)0, acc1, false, false);
            }

            const int kbase = (t << 4) + lmod;
            #pragma unroll
            for (int r = 0; r < 8; ++r) {
                float s0v = acc0[r];
                float s1v = acc1[r];
                if (s0v < bs0[r]) { bs0[r] = s0v; bi0[r] = kbase; }
                if (s1v < bs1[r]) { bs1[r] = s1v; bi1[r] = kbase; }
            }
        }

        // ---- cross-lane argmin within each 16-lane half ------------------
        #pragma unroll
        for (int r = 0; r < 8; ++r) {
            float s  = bs0[r];
            int   ki = bi0[r];
            float s2 = bs1[r];
            int   k2 = bi1[r];
            #pragma unroll
            for (int off = 8; off >= 1; off >>= 1) {
                float os = __shfl_xor(s, off, 32);
                int   oi = __shfl_xor(ki, off, 32);
                if (os < s || (os == s && oi < ki)) { s = os; ki = oi; }
                float os2 = __shfl_xor(s2, off, 32);
                int   oi2 = __shfl_xor(k2, off, 32);
                if (os2 < s2 || (os2 == s2 && oi2 < k2)) { s2 = os2; k2 = oi2; }
            }
            if (lmod == r) {
                out[n0 + (lhi << 3) + r]      = ki;
                out[n0 + 16 + (lhi << 3) + r] = k2;
            }
        }
    }
}

// ---------------------------------------------------------------------------
extern "C" void kernel_launch(void* const* d_in, const int* in_sizes, int n_in,
                              void* d_out, int out_size, void* d_ws, size_t ws_size,
                              hipStream_t stream) {
    (void)in_sizes; (void)n_in; (void)out_size; (void)ws_size;

    const float* z   = (const float*)d_in[0];   // [8,128,32,32,32]
    const float* emb = (const float*)d_in[1];   // [512,128]
    int*   out = (int*)d_out;                   // [8,32,32,32] int32 indices
    float* e2  = (float*)d_ws;                  // 512 floats scratch

    vq_e2_kernel<<<2, 256, 0, stream>>>(emb, e2);

    const int blocks = NROWS / 512;             // 512 rows per block
    const size_t lds_bytes = 32 * NCHUNK * 32 * sizeof(v2f);   // 264 KB
    vq_argmin_kernel<<<blocks, 256, lds_bytes, stream>>>(z, emb, e2, out);
}